// CIFARDiffusionLayer_70119636074792
// MI455X (gfx1250) — compile-verified
//
#include <hip/hip_runtime.h>

// ---------------------------------------------------------------------------
// ADI diffusion (CIFARDiffusionLayer): B=128, C=3, S=256, 3 steps.
// One workgroup owns one (b,c) 256x256 image, resident in CDNA5's 320KB LDS
// (stride-258 padding -> 264KB) across all 3 ADI steps. Global traffic =
// 1 read + 1 write of u (~200MB) -> HBM roofline (~9us at 23.3 TB/s).
// Image <-> LDS movement uses the Tensor Data Mover (tensor_load_to_lds /
// tensor_store_from_lds): tile_dim0=258 over tensor_dim0=256/stride 256 makes
// the TDM produce/consume the padded LDS layout directly (OOB reads -> zero
// pad cells, OOB writes dropped). Thomas coefficients are data-independent and
// precomputed once into workspace.
// ---------------------------------------------------------------------------

#define SDIM 256
#define LSTRIDE 258          // pad 2: conflict-free LDS banks both directions
#define EPSF 1e-6f

typedef int v2i __attribute__((ext_vector_type(2)));
typedef unsigned u32x4 __attribute__((ext_vector_type(4)));
typedef int i32x4 __attribute__((ext_vector_type(4)));
typedef int i32x8 __attribute__((ext_vector_type(8)));

// ---- CDNA5 feature detection (guarded; plain copies as fallback) ----
#if defined(__gfx1250__) && __has_builtin(__builtin_amdgcn_tensor_load_to_lds) && \
    __has_builtin(__builtin_amdgcn_tensor_store_from_lds)
#define HAVE_TDM 1
#else
#define HAVE_TDM 0
#endif
#if defined(__gfx1250__) && __has_builtin(__builtin_amdgcn_global_load_async_to_lds_b64)
#define HAVE_ASYNC_LOAD 1
#else
#define HAVE_ASYNC_LOAD 0
#endif
#if defined(__gfx1250__) && __has_builtin(__builtin_amdgcn_global_store_async_from_lds_b64)
#define HAVE_ASYNC_STORE 1
#else
#define HAVE_ASYNC_STORE 0
#endif

__device__ __forceinline__ void wait_async_zero() {
#if defined(__gfx1250__)
#if __has_builtin(__builtin_amdgcn_s_wait_asynccnt)
  __builtin_amdgcn_s_wait_asynccnt(0);
#else
  asm volatile("s_wait_asynccnt 0" ::: "memory");
#endif
#endif
}

__device__ __forceinline__ void wait_tensor_zero() {
#if defined(__gfx1250__)
#if __has_builtin(__builtin_amdgcn_s_wait_tensorcnt)
  __builtin_amdgcn_s_wait_tensorcnt(0);
#else
  asm volatile("s_wait_tensorcnt 0" ::: "memory");
#endif
#endif
}

__device__ __forceinline__ void async_load_b64(const float* gsrc, float* ldst) {
#if HAVE_ASYNC_LOAD
  __builtin_amdgcn_global_load_async_to_lds_b64(
      (__attribute__((address_space(1))) v2i*)gsrc,
      (__attribute__((address_space(3))) v2i*)ldst, 0, 0);
#else
  *(float2*)ldst = *(const float2*)gsrc;
#endif
}

__device__ __forceinline__ void async_store_b64(float* gdst, const float* lsrc) {
#if HAVE_ASYNC_STORE
  __builtin_amdgcn_global_store_async_from_lds_b64(
      (__attribute__((address_space(1))) v2i*)gdst,
      (__attribute__((address_space(3))) v2i*)lsrc, 0, 0);
#else
  *(float2*)gdst = *(const float2*)lsrc;
#endif
}

// ---------------------------------------------------------------------------
// Main ADI solver. One block per (b,c) image; 256 threads (8 waves).
// ---------------------------------------------------------------------------
__global__ __launch_bounds__(256) void k_adi(const float* __restrict__ u_in,
                                             const float* __restrict__ coupling,
                                             const float* __restrict__ wsr,
                                             const float* __restrict__ wst,
                                             float* __restrict__ out) {
  __shared__ __align__(16) float cpa[SDIM];
  __shared__ __align__(16) float qa[SDIM];
  __shared__ __align__(16) float cpb[SDIM];
  __shared__ __align__(16) float qb[SDIM];
  __shared__ __align__(16) float sm[SDIM * LSTRIDE];

  const int ib = blockIdx.x;            // b*3 + c
  const int tid = threadIdx.x;
  const int c = ib % 3;
  const float* g = u_in + (size_t)ib * (SDIM * SDIM);
  float* go = out + (size_t)ib * (SDIM * SDIM);

#if HAVE_TDM
  // ---- D# group1: data_size=4B, tile 258x256, tensor 256-wide stride 256 ----
  i32x8 d1;
  d1[0] = (int)(2u << 16);          // workgroup_mask=0 | data_size=2(4B) | no flags
  d1[1] = (int)(256u << 16);        // tensor_dim0[15:0]=256 (atomic_barrier_addr=0)
  d1[2] = (int)(256u << 16);        // tensor_dim0[31:16]=0 | tensor_dim1[15:0]=256
  d1[3] = (int)(258u << 16);        // tensor_dim1[31:16]=0 | tile_dim0=258
  d1[4] = 256;                      // tile_dim1=256 | tile_dim2=0
  d1[5] = 256;                      // tensor_dim0_stride[31:0]=256
  d1[6] = 0;                        // tensor_dim0_stride[47:32]=0 | dim1_stride lo=0
  d1[7] = 0;
  const i32x4 dz = {0, 0, 0, 0};    // groups 2/3: 2D tensor, unused dims zero
  const i32x8 dz8 = {0, 0, 0, 0, 0, 0, 0, 0};
  const unsigned lds_off =
      (unsigned)(unsigned long long)(__attribute__((address_space(3))) float*)sm;

  // ---- load: one TDM DMA of the whole tile (issued by wave 0 only) ----
  if (tid == 0) {
    const unsigned long long ga = (unsigned long long)(__UINTPTR_TYPE__)g;
    u32x4 d0;
    d0.x = 1u;                                     // count=1 valid descriptor
    d0.y = lds_off;                                // lds_addr (bytes)
    d0.z = (unsigned)(ga & 0xffffffffull);         // global_addr[31:0]
    d0.w = (unsigned)((ga >> 32) & 0x01ffffffull)  // global_addr[56:32]
           | (2u << 30);                           // type=2 (image)
    __builtin_amdgcn_tensor_load_to_lds(d0, d1, dz, dz, dz8, 0);
  }
  wait_tensor_zero();
#else
  for (int k = 0; k < 128; ++k) {
    const int e = (k * 256 + tid) * 2;
    const int row = e >> 8;
    const int col = e & 255;
    async_load_b64(g + e, sm + row * LSTRIDE + col);
  }
  wait_async_zero();
#endif
  __syncthreads();

  const float dscale = coupling[c * 3 + c];  // diagonal of channel_coupling

  for (int step = 0; step < 3; ++step) {
    const int sa = c * 3 + step;        // alpha sequence id
    const int sb = 9 + c * 3 + step;    // beta sequence id
    // stage coefficient tables into LDS (broadcast-friendly)
    cpa[tid] = wst[sa * 512 + tid];
    qa[tid] = wst[sa * 512 + 256 + tid];
    cpb[tid] = wst[sb * 512 + tid];
    qb[tid] = wst[sb * 512 + 256 + tid];
    const float r_a = wsr[sa];
    const float r_b = wsr[sb];
    __syncthreads();

    // ---------------- X sweep: thread tid owns row tid ----------------
    {
      float* row = sm + tid * LSTRIDE;
      float dp = row[0] * qa[0];
      row[0] = dp;
      dp = (row[1] + r_a * dp) * qa[1];
      row[1] = dp;
#pragma unroll 4
      for (int i = 2; i < SDIM; i += 2) {
        const float2 d = *(const float2*)(row + i);
        const float2 qv = *(const float2*)(qa + i);
        const float d0 = (d.x + r_a * dp) * qv.x;
        const float d1 = (d.y + r_a * d0) * qv.y;
        *(float2*)(row + i) = make_float2(d0, d1);
        dp = d1;
      }
      // back substitution (x[255] = dp[255], already in place)
      float x = dp;
      x = row[254] - cpa[254] * x;
      row[254] = x;
#pragma unroll 4
      for (int i = 252; i >= 0; i -= 2) {
        const float2 d = *(const float2*)(row + i);
        const float2 cv = *(const float2*)(cpa + i);
        const float x1 = d.y - cv.y * x;
        const float x0 = d.x - cv.x * x1;
        *(float2*)(row + i) = make_float2(x0, x1);
        x = x0;
      }
    }
    __syncthreads();

    // ------- Y sweep: thread tid owns column tid; fuse diag scale -------
    {
      float* colp = sm + tid;
      float dp = colp[0] * qb[0];
      colp[0] = dp;
#pragma unroll 4
      for (int i = 1; i < SDIM; ++i) {
        dp = (colp[i * LSTRIDE] + r_b * dp) * qb[i];
        colp[i * LSTRIDE] = dp;
      }
      float x = dp;                               // x[255]
      colp[(SDIM - 1) * LSTRIDE] = x * dscale;    // store scaled, carry unscaled
#pragma unroll 4
      for (int i = SDIM - 2; i >= 0; --i) {
        const float xn = colp[i * LSTRIDE] - cpb[i] * x;
        colp[i * LSTRIDE] = xn * dscale;
        x = xn;
      }
    }
    __syncthreads();
  }

  // ---- store: padded LDS -> global ----
#if HAVE_TDM
  if (tid == 0) {
    const unsigned long long ga = (unsigned long long)(__UINTPTR_TYPE__)go;
    u32x4 d0;
    d0.x = 1u;
    d0.y = lds_off;
    d0.z = (unsigned)(ga & 0xffffffffull);
    d0.w = (unsigned)((ga >> 32) & 0x01ffffffull) | (2u << 30);
    // same tile 258 x 256: the 2 pad lanes per row are OOB -> writes dropped
    __builtin_amdgcn_tensor_store_from_lds(d0, d1, dz, dz, dz8, 0);
  }
  wait_tensor_zero();   // keep LDS alive until DMA read completes
#else
  for (int k = 0; k < 128; ++k) {
    const int e = (k * 256 + tid) * 2;
    const int row = e >> 8;
    const int col = e & 255;
    async_store_b64(go + e, sm + row * LSTRIDE + col);
  }
  wait_async_zero();
#endif
}

// ---------------------------------------------------------------------------
// Kernel 1: per-channel spatial means of alpha_spatial / beta_spatial.
// blockIdx 0..2 -> alpha channel, 3..5 -> beta channel. 256 threads.
// ---------------------------------------------------------------------------
__global__ __launch_bounds__(256) void k_means(const float* __restrict__ alpha_sp,
                                               const float* __restrict__ beta_sp,
                                               float* __restrict__ means) {
  __shared__ float red[256];
  const int which = blockIdx.x;
  const float* src = (which < 3) ? (alpha_sp + which * (SDIM * SDIM))
                                 : (beta_sp + (which - 3) * (SDIM * SDIM));
  float acc = 0.0f;
  for (int k = 0; k < SDIM; ++k) acc += src[k * SDIM + threadIdx.x];
  red[threadIdx.x] = acc;
  __syncthreads();
  for (int off = 128; off > 0; off >>= 1) {
    if ((int)threadIdx.x < off) red[threadIdx.x] += red[threadIdx.x + off];
    __syncthreads();
  }
  if (threadIdx.x == 0) means[which] = red[0] * (1.0f / (SDIM * SDIM));
}

// ---------------------------------------------------------------------------
// Kernel 2: build Thomas tables. 18 sequences: dir(2) x channel(3) x step(3).
// Faithful to reference: cp0 = -r/(1+r+eps); interior denom clamped to eps;
// b[last] = 1+r. Stores cp[256] then q[256]=1/denom per sequence, plus r.
// ---------------------------------------------------------------------------
__global__ void k_tables(const float* __restrict__ alpha_base,
                         const float* __restrict__ beta_base,
                         const float* __restrict__ means,
                         float* __restrict__ wsr,   // 18 r values
                         float* __restrict__ wst) { // 18 * 512 floats
  const int s = threadIdx.x;
  if (s >= 18) return;
  const int dir = s / 9;          // 0 = alpha (x), 1 = beta (y)
  const int c = (s % 9) / 3;
  const int step = s % 3;
  const float base = dir ? beta_base[c] : alpha_base[c];
  const float mean = means[dir * 3 + c];
  const float t = (float)step * 0.05f;                       // step * DT
  const float coef = fminf(fmaxf(base + mean * t, EPSF), 1.0f);
  const float r = coef * 0.025f;                             // (DT/2)/DX^2
  wsr[s] = r;
  float* cp = wst + s * 512;
  float* q = cp + 256;
  float den = 1.0f + r + EPSF;    // b[0] + eps (no clamp on first)
  float cprev = -r / den;
  cp[0] = cprev;
  q[0] = 1.0f / den;
  for (int i = 1; i < SDIM; ++i) {
    const float bi = (i == SDIM - 1) ? (1.0f + r) : (1.0f + 2.0f * r);
    const float d = fmaxf(bi + r * cprev, EPSF);  // b - a*cp, a = -r
    const float ci = -r / d;
    cp[i] = ci;
    q[i] = 1.0f / d;
    cprev = ci;
  }
}

// ---------------------------------------------------------------------------
// Host launcher
// ---------------------------------------------------------------------------
extern "C" void kernel_launch(void* const* d_in, const int* in_sizes, int n_in,
                              void* d_out, int out_size, void* d_ws, size_t ws_size,
                              hipStream_t stream) {
  const float* u = (const float*)d_in[0];
  const float* alpha_base = (const float*)d_in[1];
  const float* beta_base = (const float*)d_in[2];
  const float* alpha_sp = (const float*)d_in[3];
  const float* beta_sp = (const float*)d_in[4];
  const float* coupling = (const float*)d_in[5];
  float* out = (float*)d_out;

  float* ws = (float*)d_ws;
  float* means = ws;        // 6 floats
  float* wsr = ws + 8;      // 18 floats
  float* wst = ws + 32;     // 18*512 floats

  const int nimg = in_sizes[0] / (SDIM * SDIM);  // B*C = 384

  hipLaunchKernelGGL(k_means, dim3(6), dim3(256), 0, stream, alpha_sp, beta_sp, means);
  hipLaunchKernelGGL(k_tables, dim3(1), dim3(32), 0, stream, alpha_base, beta_base,
                     means, wsr, wst);
  hipLaunchKernelGGL(k_adi, dim3(nimg), dim3(256), 0, stream, u, coupling, wsr, wst, out);
}